// Net_60163901882525
// MI455X (gfx1250) — compile-verified
//
#include <hip/hip_runtime.h>
#include <hip/hip_bf16.h>
#include <math.h>

// GCN: 5-layer, N=10000 nodes, E=320000 edges, F_IN=H=256, C=40.
// d_out = [log_softmax(logits) (N*C) | logits (N*C)], all fp32.

#define GN 10000
#define GE 320000
#define GF 256
#define GH 256
#define GC 40
#define KDIM 256
#define WPAD 48      // W5 padded column count (3 tiles of 16)

typedef float v2f __attribute__((ext_vector_type(2)));
typedef float v8f __attribute__((ext_vector_type(8)));
typedef unsigned int v4u __attribute__((ext_vector_type(4)));
typedef int v4i __attribute__((ext_vector_type(4)));
typedef int v8i __attribute__((ext_vector_type(8)));

#if defined(__has_builtin)
#if __has_builtin(__builtin_amdgcn_tensor_load_to_lds) && \
    __has_builtin(__builtin_amdgcn_s_wait_tensorcnt)
#define USE_TDM 1
#endif
#endif

// ---------------- utility kernels ----------------

__global__ void fill_kernel(float* __restrict__ p, float v, int n) {
    int i = blockIdx.x * blockDim.x + threadIdx.x;
    if (i < n) p[i] = v;
}

// deg[dst] += 1 for each edge; deg pre-filled with 1.0 (self loop).
__global__ void degree_kernel(const int* __restrict__ ei, float* __restrict__ deg, int E) {
    int e = blockIdx.x * blockDim.x + threadIdx.x;
    if (e < E) atomicAdd(&deg[ei[E + e]], 1.0f);
}

__global__ void dinv_kernel(const float* __restrict__ deg, float* __restrict__ dinv, int n) {
    int i = blockIdx.x * blockDim.x + threadIdx.x;
    if (i < n) {
        float d = deg[i];
        dinv[i] = (d > 0.0f) ? rsqrtf(d) : 0.0f;
    }
}

// W5 [256 x 40] -> Wpad [256 x 48], zero padded.
__global__ void pad_w5_kernel(const float* __restrict__ W5, float* __restrict__ Wp) {
    int i = blockIdx.x * blockDim.x + threadIdx.x;
    if (i >= KDIM * WPAD) return;
    int k = i / WPAD, n = i - k * WPAD;
    Wp[i] = (n < GC) ? W5[k * GC + n] : 0.0f;
}

// ---------------- WMMA fp32 GEMM: out[nrows x dout] = A[nrows x KDIM] @ W[KDIM x wstride]
// Block = 256 threads = 8 waves; one 16-col N-tile, 8 consecutive M-tiles.
// W slab [KDIM x 16] staged into LDS (16 KB) by the Tensor Data Mover (wave 0),
// then per-wave branch-free K loop with dual accumulators:
//   64x V_WMMA_F32_16X16X4_F32 per wave, A via 8B global loads, B via ds_load.
__global__ void gemm_wmma_f32(const float* __restrict__ A, const float* __restrict__ W,
                              float* __restrict__ out, int mtiles, int ntiles,
                              int wstride, int dout) {
    __shared__ float wlds[KDIM * 16];

    int nt = blockIdx.x % ntiles;
    int bm = blockIdx.x / ntiles;
    int nb = nt * 16;

#ifdef USE_TDM
    // --- TDM: DMA the [KDIM x 16] strided slab of W into contiguous LDS ---
    if (threadIdx.x < 32) {                      // wave 0 issues the descriptor
        unsigned long long gaddr = (unsigned long long)(const void*)(W + nb);
        unsigned ldsoff = (unsigned)(unsigned long long)(void*)wlds;
        // D# group 0: count=1 | lds_addr | global_addr[56:0] | type=2
        v4u g0 = { 1u,
                   ldsoff,
                   (unsigned)(gaddr & 0xffffffffu),
                   (unsigned)((gaddr >> 32) & 0x01ffffffu) | 0x80000000u };
        // D# group 1: data_size=4B(code 2); tensor_dim0=wstride; tensor_dim1=KDIM;
        // tile_dim0=16; tile_dim1=KDIM; tensor_dim0_stride=wstride; rest 0.
        unsigned ws_u = (unsigned)wstride;
        v8i g1 = { (int)0x00020000u,                               // data_size=2
                   (int)((ws_u & 0xffffu) << 16),                  // tensor_dim0 lo
                   (int)((ws_u >> 16) | ((unsigned)KDIM << 16)),   // td0 hi | td1 lo
                   (int)(((unsigned)KDIM >> 16) | (16u << 16)),    // td1 hi | tile_dim0
                   (int)KDIM,                                      // tile_dim1 (tile_dim2=0)
                   (int)ws_u,                                      // tensor_dim0_stride lo
                   0, 0 };
        v4i gz = { 0, 0, 0, 0 };
#if __clang_major__ >= 23
        v8i gz8 = { 0, 0, 0, 0, 0, 0, 0, 0 };
        __builtin_amdgcn_tensor_load_to_lds(g0, g1, gz, gz, gz8, 0);
#else
        __builtin_amdgcn_tensor_load_to_lds(g0, g1, gz, gz, 0);
#endif
        __builtin_amdgcn_s_wait_tensorcnt(0);
    }
    __syncthreads();
#else
    // --- fallback: cooperative load of W slab: [k][n], n contiguous ---
    for (int i = threadIdx.x; i < KDIM * 16; i += 256) {
        int k = i >> 4, n = i & 15;
        wlds[i] = W[k * wstride + nb + n];
    }
    __syncthreads();
#endif

    int wave = threadIdx.x >> 5;
    int lane = threadIdx.x & 31;
    int mtraw = bm * 8 + wave;
    int mt = (mtraw < mtiles) ? mtraw : (mtiles - 1);   // clamp: EXEC stays full
    int mb = mt * 16;

    int m    = lane & 15;
    int koff = (lane >> 4) << 1;     // 0 / 2
    int n    = lane & 15;

    const float* __restrict__ arow = A + (long long)(mb + m) * KDIM;

    v8f acc0 = {}, acc1 = {};
    #pragma unroll 4
    for (int k0 = 0; k0 < KDIM; k0 += 8) {
        v2f a0 = *(const v2f*)(arow + k0 + koff);
        v2f a1 = *(const v2f*)(arow + k0 + 4 + koff);
        v2f b0, b1;
        b0.x = wlds[(k0 + koff) * 16 + n];
        b0.y = wlds[(k0 + koff + 1) * 16 + n];
        b1.x = wlds[(k0 + 4 + koff) * 16 + n];
        b1.y = wlds[(k0 + 5 + koff) * 16 + n];
        acc0 = __builtin_amdgcn_wmma_f32_16x16x4_f32(
            false, a0, false, b0, (short)0, acc0, false, false);
        acc1 = __builtin_amdgcn_wmma_f32_16x16x4_f32(
            false, a1, false, b1, (short)0, acc1, false, false);
    }
    v8f acc = acc0 + acc1;

    int col = nb + n;
    if (mtraw < mtiles && col < dout) {
        int mo = (lane >> 4) << 3;   // +8 rows for lanes 16-31
        #pragma unroll
        for (int v = 0; v < 8; ++v) {
            out[(long long)(mb + v + mo) * dout + col] = acc[v];
        }
    }
}

// ---------------- edge scatter: out[dst] += in[src] * dinv[src]*dinv[dst] -------------
// One wave per logical edge; e in [E, E+N) are self loops.
__global__ void scatter256_kernel(const float* __restrict__ in, float* __restrict__ out,
                                  const int* __restrict__ ei, const float* __restrict__ dinv,
                                  int E, int N) {
    int wid  = (blockIdx.x * blockDim.x + threadIdx.x) >> 5;
    int lane = threadIdx.x & 31;
    if (wid >= E + N) return;
    int s, d; float w;
    if (wid < E) {
        s = ei[wid]; d = ei[E + wid];
        w = dinv[s] * dinv[d];
    } else {
        s = d = wid - E;
        w = dinv[s] * dinv[s];
    }
    const float4* __restrict__ in4 = (const float4*)(in + (long long)s * 256);
    float* __restrict__ orow = out + (long long)d * 256;
    #pragma unroll
    for (int j = 0; j < 2; ++j) {
        float4 v = in4[lane + 32 * j];          // coalesced b128
        int f = (lane + 32 * j) * 4;
        atomicAdd(orow + f + 0, v.x * w);
        atomicAdd(orow + f + 1, v.y * w);
        atomicAdd(orow + f + 2, v.z * w);
        atomicAdd(orow + f + 3, v.w * w);
    }
}

__global__ void scatter40_kernel(const float* __restrict__ in, float* __restrict__ out,
                                 const int* __restrict__ ei, const float* __restrict__ dinv,
                                 int E, int N) {
    int wid  = (blockIdx.x * blockDim.x + threadIdx.x) >> 5;
    int lane = threadIdx.x & 31;
    if (wid >= E + N) return;
    int s, d; float w;
    if (wid < E) {
        s = ei[wid]; d = ei[E + wid];
        w = dinv[s] * dinv[d];
    } else {
        s = d = wid - E;
        w = dinv[s] * dinv[s];
    }
    const float* __restrict__ irow = in + (long long)s * GC;
    float* __restrict__ orow = out + (long long)d * GC;
    #pragma unroll
    for (int j = 0; j < 2; ++j) {
        int f = lane + 32 * j;
        if (f < GC) atomicAdd(orow + f, irow[f] * w);
    }
}

// ---------------- bias (+ optional relu), in place ----------------
__global__ void bias_act_kernel(float* __restrict__ buf, const float* __restrict__ bias,
                                int n, int dout, int do_relu) {
    int gid = blockIdx.x * blockDim.x + threadIdx.x;
    if (gid >= n) return;
    int f = gid % dout;
    float v = buf[gid] + bias[f];
    buf[gid] = do_relu ? fmaxf(v, 0.0f) : v;
}

// ---------------- log-softmax over C classes; emits [logp | logits] ----------------
__global__ void logsoftmax_kernel(const float* __restrict__ logits,
                                  float* __restrict__ out, int n, int C) {
    int i = blockIdx.x * blockDim.x + threadIdx.x;
    if (i >= n) return;
    const float* row = logits + (long long)i * C;
    float mx = -INFINITY;
    for (int c = 0; c < C; ++c) mx = fmaxf(mx, row[c]);
    float s = 0.0f;
    for (int c = 0; c < C; ++c) s += __expf(row[c] - mx);
    float lse = mx + __logf(s);
    float* logp = out + (long long)i * C;
    float* lcpy = out + (long long)n * C + (long long)i * C;
    for (int c = 0; c < C; ++c) {
        float l = row[c];
        logp[c] = l - lse;
        lcpy[c] = l;
    }
}

// ---------------- driver ----------------

static inline int cdiv(long long a, long long b) { return (int)((a + b - 1) / b); }

static void run_layer(const float* h_in, const float* Wm, const float* bm,
                      float* gbuf, float* obuf, const int* ei, const float* dinv,
                      int wstride, int dout, int do_relu, hipStream_t stream) {
    int ntiles = wstride / 16;
    int mtiles = GN / 16;                         // 625, exact
    int blocks = cdiv(mtiles, 8) * ntiles;        // 8 M-tiles per block
    gemm_wmma_f32<<<blocks, 256, 0, stream>>>(h_in, Wm, gbuf, mtiles, ntiles,
                                              wstride, dout);
    long long nelem = (long long)GN * dout;
    fill_kernel<<<cdiv(nelem, 256), 256, 0, stream>>>(obuf, 0.0f, (int)nelem);
    int swaves = GE + GN;                         // one wave per logical edge
    if (dout == 256)
        scatter256_kernel<<<cdiv((long long)swaves * 32, 256), 256, 0, stream>>>(
            gbuf, obuf, ei, dinv, GE, GN);
    else
        scatter40_kernel<<<cdiv((long long)swaves * 32, 256), 256, 0, stream>>>(
            gbuf, obuf, ei, dinv, GE, GN);
    bias_act_kernel<<<cdiv(nelem, 256), 256, 0, stream>>>(
        obuf, bm, (int)nelem, dout, do_relu);
}

extern "C" void kernel_launch(void* const* d_in, const int* in_sizes, int n_in,
                              void* d_out, int out_size, void* d_ws, size_t ws_size,
                              hipStream_t stream) {
    const float* x  = (const float*)d_in[0];
    const int*   ei = (const int*)d_in[1];       // [2, E]: row 0 = src, row 1 = dst
    const float* W1 = (const float*)d_in[2];
    const float* b1 = (const float*)d_in[3];
    const float* W2 = (const float*)d_in[4];
    const float* b2 = (const float*)d_in[5];
    const float* W3 = (const float*)d_in[6];
    const float* b3 = (const float*)d_in[7];
    const float* W4 = (const float*)d_in[8];
    const float* b4 = (const float*)d_in[9];
    const float* W5 = (const float*)d_in[10];
    const float* b5 = (const float*)d_in[11];
    (void)in_sizes; (void)n_in; (void)out_size; (void)ws_size;

    // Workspace (floats): deg[N] | dinv[N] | G[N*H] | buf0[N*H] | buf1[N*H] | Wpad[256*48]
    float* ws   = (float*)d_ws;
    float* deg  = ws;
    float* dinv = ws + GN;
    float* G    = ws + 2 * GN;
    float* buf0 = G    + (long long)GN * GH;
    float* buf1 = buf0 + (long long)GN * GH;
    float* Wpad = buf1 + (long long)GN * GH;

    // GCN normalization
    fill_kernel<<<cdiv(GN, 256), 256, 0, stream>>>(deg, 1.0f, GN);   // self loops
    degree_kernel<<<cdiv(GE, 256), 256, 0, stream>>>(ei, deg, GE);
    dinv_kernel<<<cdiv(GN, 256), 256, 0, stream>>>(deg, dinv, GN);
    pad_w5_kernel<<<cdiv(KDIM * WPAD, 256), 256, 0, stream>>>(W5, Wpad);

    // Layers
    run_layer(x,    W1,   b1, G, buf0, ei, dinv, GH,   GH, 1, stream);
    run_layer(buf0, W2,   b2, G, buf1, ei, dinv, GH,   GH, 1, stream);
    run_layer(buf1, W3,   b3, G, buf0, ei, dinv, GH,   GH, 1, stream);
    run_layer(buf0, W4,   b4, G, buf1, ei, dinv, GH,   GH, 1, stream);
    run_layer(buf1, Wpad, b5, G, buf0, ei, dinv, WPAD, GC, 0, stream);  // logits in buf0

    // log_softmax + copy logits: d_out = [logp (N*C) | logits (N*C)]
    logsoftmax_kernel<<<cdiv(GN, 256), 256, 0, stream>>>(buf0, (float*)d_out, GN, GC);
}